// OptimizedS4Layer_24352464569722
// MI455X (gfx1250) — compile-verified
//
#include <hip/hip_runtime.h>
#include <math.h>
#include <stdint.h>

// ---------------- problem constants (from reference) ----------------
#define S_LEN 4096
#define D_DIM 256
#define N_DIM 16
#define B_DIM 16
#define DT_F  1e-4f
#define LN_EPS_F 1e-5f

typedef float v2f __attribute__((ext_vector_type(2)));
typedef float v8f __attribute__((ext_vector_type(8)));
typedef unsigned int u32x4 __attribute__((ext_vector_type(4)));
typedef int i32x4 __attribute__((ext_vector_type(4)));
typedef int i32x8 __attribute__((ext_vector_type(8)));

// =====================================================================
// TDM: DMA one 16x256 f32 tile (global, row-stride 256 floats) into LDS
// laid out as [16][260] floats (4-DWORD pad per 256-DWORD row to kill
// bank conflicts). One tensor_load_to_lds per wave, TENSORcnt-tracked.
// D# bitfields per CDNA5 ISA.
// This toolchain uses the 6-arg builtin (clang-23 / therock-10.0 form).
// =====================================================================
__device__ __forceinline__ void tdm_load_tile_16x256(const float* gsrc,
                                                     unsigned lds_byte_off)
{
    const unsigned long long ga = (unsigned long long)(uintptr_t)gsrc;
    const unsigned ga_lo = __builtin_amdgcn_readfirstlane((unsigned)ga);
    const unsigned ga_hi = __builtin_amdgcn_readfirstlane((unsigned)(ga >> 32));
    const unsigned ldsb  = __builtin_amdgcn_readfirstlane(lds_byte_off);

    u32x4 g0;
    g0.x = 1u;                          // count=1 (valid), is_restore=0, no gather
    g0.y = ldsb;                        // lds_addr        -> bits [63:32]
    g0.z = ga_lo;                       // global_addr lo  -> bits [95:64]
    g0.w = (ga_hi & 0x01FFFFFFu)        // global_addr[56:32] -> bits [120:96]
         | (2u << 30);                  // type=2 ("image")   -> bits [127:126]

    i32x8 g1;
    g1[0] = (2 << 16)                   // data_size=2 (4 bytes), workgroup_mask=0
          | (1 << 20)                   // pad_enable
          | (7 << 22)                   // pad_interval: 256 DWORDs
          | (3 << 25);                  // pad_amount:   4 DWORDs
    g1[1] = (256 << 16);                // tensor_dim0[15:0]=256 at bits [63:48]
    g1[2] = (16  << 16);                // tensor_dim0 hi=0 | tensor_dim1[15:0]=16
    g1[3] = (256 << 16);                // tensor_dim1 hi=0 | tile_dim0=256
    g1[4] = 16;                         // tile_dim1=16, tile_dim2=0
    g1[5] = 256;                        // tensor_dim0_stride = 256 elements
    g1[6] = 0;                          // stride0 hi | tensor_dim1_stride lo
    g1[7] = 0;

    const i32x4 z4 = {0, 0, 0, 0};      // groups 2/3 unused (2-D tensor)
    const i32x8 z8 = {0, 0, 0, 0, 0, 0, 0, 0};
    __builtin_amdgcn_tensor_load_to_lds(g0, g1, z4, z4, z8, 0);
}

// =====================================================================
// Phase 1: V[s*256 + b*16 + n] = sum_d x[b,s,d] * (DT*Bm[n,d])
// One wave = one 16-row x 16-col tile, K=256 via 64 chained
// v_wmma_f32_16x16x4_f32.  x-tile arrives via TDM. blockDim=64.
// =====================================================================
__global__ void __launch_bounds__(64)
s4_phase1_xB(const float* __restrict__ x,
             const float* __restrict__ Bm,
             float* __restrict__ V)
{
    __shared__ float ldsB[N_DIM][260];       // DT*Bm, padded
    __shared__ float ldsX[2][16][260];       // per-wave x tile (TDM dest)

    const int tid  = threadIdx.x;            // 0..63
    const int wave = tid >> 5;               // 0..1
    const int lane = tid & 31;
    const int hi   = lane >> 4;              // K-pair select
    const int lo   = lane & 15;              // M (A) / N (B)

    const int  tile = blockIdx.x * 2 + wave;         // 0..4095
    const long r0   = (long)tile * 16;               // first row (b*S + s)

    // async DMA the 16x256 x-tile straight into (padded) LDS
    tdm_load_tile_16x256(x + r0 * D_DIM,
                         (unsigned)(uintptr_t)&ldsX[wave][0][0]);

    // stage DT*Bm (16x256) cooperatively through VGPRs meanwhile
    for (int t = tid; t < N_DIM * D_DIM; t += 64) {
        ldsB[t >> 8][t & 255] = DT_F * Bm[t];
    }

    __builtin_amdgcn_s_wait_tensorcnt(0);    // own tile resident
    __syncthreads();                         // ldsB visible to both waves

    v8f acc = {};
    #pragma unroll 8
    for (int k0 = 0; k0 < D_DIM; k0 += 4) {
        const int k = k0 + 2 * hi;
        v2f a; a.x = ldsX[wave][lo][k]; a.y = ldsX[wave][lo][k + 1];  // A[m][k]
        v2f b; b.x = ldsB[lo][k];       b.y = ldsB[lo][k + 1];        // B[k][n]
        acc = __builtin_amdgcn_wmma_f32_16x16x4_f32(
                  false, a, false, b, (short)0, acc, false, false);
    }

    const int b_idx = (int)(r0 >> 12);
    const int s0    = (int)(r0 & 4095);
    #pragma unroll
    for (int j = 0; j < 8; ++j) {
        const int m = j + 8 * hi;            // lane holds D[m][n=lo] in elem j
        V[(size_t)(s0 + m) * 256 + b_idx * 16 + lo] = acc[j];
    }
}

// =====================================================================
// Phase 2: the only serial part. 256 independent scalar recurrences,
// one lane per (b,n):  h = clip(h*a_d + V[s]),  H[s] = h.
// =====================================================================
__global__ void __launch_bounds__(256)
s4_phase2_scan(const float* __restrict__ V,
               const float* __restrict__ A,
               float* __restrict__ H)
{
    const int tid = threadIdx.x;                  // 0..255 == b*16+n
    const float a_d = __expf(-DT_F * fabsf(A[tid & 15]));
    float h = 0.0f;

    #pragma unroll 4
    for (int s = 0; s < S_LEN; ++s) {
        if (s + 32 < S_LEN)
            __builtin_prefetch(&V[(size_t)(s + 32) * 256 + tid], 0, 0);
        const float v = V[(size_t)s * 256 + tid];
        h = fminf(10.0f, fmaxf(-10.0f, fmaf(h, a_d, v)));
        H[(size_t)s * 256 + tid] = h;
    }
}

// =====================================================================
// Phase 3: y = clip(H @ Cm^T + u*Dv), fused LayerNorm over D=256.
// A (16x16 H-tile) in 4 v2f regs reused across 16 D-chunks; 64 WMMAs
// per tile. x-tile via TDM. Row stats via __shfl_xor in 16-lane halves.
// =====================================================================
__global__ void __launch_bounds__(64)
s4_phase3_yC_ln(const float* __restrict__ x,
                const float* __restrict__ H,
                const float* __restrict__ Cm,
                const float* __restrict__ Dv,
                const float* __restrict__ gamma,
                const float* __restrict__ beta,
                float* __restrict__ out)
{
    __shared__ float ldsC[D_DIM][18];        // Cm[d][k], padded
    __shared__ float ldsX[2][16][260];       // per-wave x tile (TDM dest)

    const int tid  = threadIdx.x;
    const int wave = tid >> 5;
    const int lane = tid & 31;
    const int hi   = lane >> 4;
    const int lo   = lane & 15;

    const int  tile  = blockIdx.x * 2 + wave;
    const long r0    = (long)tile * 16;
    const int  b_idx = (int)(r0 >> 12);
    const int  s0    = (int)(r0 & 4095);

    tdm_load_tile_16x256(x + r0 * D_DIM,
                         (unsigned)(uintptr_t)&ldsX[wave][0][0]);

    for (int t = tid; t < D_DIM * N_DIM; t += 64) {
        ldsC[t >> 4][t & 15] = Cm[t];
    }

    // A-matrix (H tile) into registers: A[m=lo][k = kk*4 + 2*hi + {0,1}]
    v2f areg[4];
    #pragma unroll
    for (int kk = 0; kk < 4; ++kk) {
        const float* hp = &H[(size_t)(s0 + lo) * 256 + b_idx * 16 + kk * 4 + 2 * hi];
        areg[kk].x = hp[0];
        areg[kk].y = hp[1];
    }

    // per-lane params for column n = lo of every 16-wide D chunk
    float dv[16], gm[16], bt[16];
    #pragma unroll
    for (int c = 0; c < 16; ++c) {
        dv[c] = Dv[c * 16 + lo];
        gm[c] = gamma[c * 16 + lo];
        bt[c] = beta[c * 16 + lo];
    }

    __builtin_amdgcn_s_wait_tensorcnt(0);
    __syncthreads();

    // 16 D-chunks x (K=16 as 4 chained WMMAs)
    v8f acc[16];
    #pragma unroll
    for (int c = 0; c < 16; ++c) {
        const int d0 = c * 16;
        v8f a_c = {};
        #pragma unroll
        for (int kk = 0; kk < 4; ++kk) {
            const int k = kk * 4 + 2 * hi;
            v2f b; b.x = ldsC[d0 + lo][k]; b.y = ldsC[d0 + lo][k + 1];
            a_c = __builtin_amdgcn_wmma_f32_16x16x4_f32(
                      false, areg[kk], false, b, (short)0, a_c, false, false);
        }
        acc[c] = a_c;
    }

    // y = clip(acc + u*Dv); accumulate row statistics
    float psum[8], psq[8];
    #pragma unroll
    for (int j = 0; j < 8; ++j) { psum[j] = 0.0f; psq[j] = 0.0f; }

    #pragma unroll
    for (int c = 0; c < 16; ++c) {
        const int d0 = c * 16;
        #pragma unroll
        for (int j = 0; j < 8; ++j) {
            const int m = j + 8 * hi;
            const float u = ldsX[wave][m][d0 + lo];
            float y = fmaf(u, dv[c], acc[c][j]);
            y = fminf(10.0f, fmaxf(-10.0f, y));
            acc[c][j] = y;
            psum[j] += y;
            psq[j]  = fmaf(y, y, psq[j]);
        }
    }

    // reduce over the 16 lanes holding row m (xor<16 stays in wave halves)
    #pragma unroll
    for (int off = 1; off < 16; off <<= 1) {
        #pragma unroll
        for (int j = 0; j < 8; ++j) {
            psum[j] += __shfl_xor(psum[j], off, 32);
            psq[j]  += __shfl_xor(psq[j],  off, 32);
        }
    }

    float mu[8], rs[8];
    #pragma unroll
    for (int j = 0; j < 8; ++j) {
        mu[j] = psum[j] * (1.0f / 256.0f);
        const float var = psq[j] * (1.0f / 256.0f) - mu[j] * mu[j];
        rs[j] = rsqrtf(var + LN_EPS_F);
    }

    #pragma unroll
    for (int c = 0; c < 16; ++c) {
        const int d0 = c * 16;
        #pragma unroll
        for (int j = 0; j < 8; ++j) {
            const int m = j + 8 * hi;
            out[(size_t)(r0 + m) * 256 + d0 + lo] =
                (acc[c][j] - mu[j]) * rs[j] * gm[c] + bt[c];
        }
    }
}

// =====================================================================
extern "C" void kernel_launch(void* const* d_in, const int* in_sizes, int n_in,
                              void* d_out, int out_size, void* d_ws, size_t ws_size,
                              hipStream_t stream)
{
    const float* x     = (const float*)d_in[0];   // [B,S,D]
    const float* A     = (const float*)d_in[1];   // [N]
    const float* Bm    = (const float*)d_in[2];   // [N,D]
    const float* Cm    = (const float*)d_in[3];   // [D,N]
    const float* Dv    = (const float*)d_in[4];   // [D]
    const float* gamma = (const float*)d_in[5];   // [D]
    const float* beta  = (const float*)d_in[6];   // [D]
    float*       out   = (float*)d_out;           // [B,S,D]

    // workspace: V then H, each [S][B*N] = 4096*256 floats (4 MB)
    float* V = (float*)d_ws;
    float* H = V + (size_t)S_LEN * 256;

    const int tiles  = (B_DIM * S_LEN) / 16;      // 4096 wave-tiles
    const int blocks = tiles / 2;                 // 2 waves per 64-thread block

    s4_phase1_xB   <<<blocks, 64, 0, stream>>>(x, Bm, V);
    s4_phase2_scan <<<1, 256, 0, stream>>>(V, A, H);
    s4_phase3_yC_ln<<<blocks, 64, 0, stream>>>(x, H, Cm, Dv, gamma, beta, out);
}